// GraphConv_20675972563283
// MI455X (gfx1250) — compile-verified
//
#include <hip/hip_runtime.h>
#include <hip/hip_bf16.h>

#define NV 20000
#define NC 300
#define NI 8000
#define NN 28300          // total nodes
#define NP 28304          // padded to multiple of 16
#define NE 400000
#define D 128
#define HSTRIDE 132       // LDS row stride (banks spread)
#define ENC_NEGINF 0x007FFFFFu

typedef __attribute__((ext_vector_type(2))) float v2f;
typedef __attribute__((ext_vector_type(8))) float v8f;

__device__ __forceinline__ v8f wmma4(v2f a, v2f b, v8f c) {
#if __has_builtin(__builtin_amdgcn_wmma_f32_16x16x4_f32)
  return __builtin_amdgcn_wmma_f32_16x16x4_f32(false, a, false, b, (short)0, c,
                                               false, false);
#else
  c[0] += a[0] * b[0];
  return c;
#endif
}

// Fused two-layer MLP: S = relu(X @ W1^T + b1) @ W2^T + b2
// One block = 256 threads = 8 waves; block handles 16 rows, wave w handles
// output columns [16w, 16w+16). Hidden 16x128 tile staged in LDS.
__global__ void __launch_bounds__(256)
mlp2_kernel(const float* __restrict__ X, const float* __restrict__ W1,
            const float* __restrict__ b1, const float* __restrict__ W2,
            const float* __restrict__ b2, float* __restrict__ S) {
  __shared__ __align__(16) float Ht[16 * HSTRIDE];
  const int lane = threadIdx.x & 31;
  const int wave = threadIdx.x >> 5;
  const int r0 = blockIdx.x * 16;
  const int n0 = wave * 16;
  const int mrow = lane & 15;   // M (for A) / N (for B,D)
  const int khalf = lane >> 4;  // K sub-pair select

  v8f acc = {};
  const float* xrow = X + (size_t)(r0 + mrow) * D + khalf * 2;
  const float* w1row = W1 + (size_t)(n0 + mrow) * D + khalf * 2;
#pragma unroll 8
  for (int k0 = 0; k0 < D; k0 += 4) {
    v2f a = *(const v2f*)(xrow + k0);
    v2f b = *(const v2f*)(w1row + k0);
    acc = wmma4(a, b, acc);
  }
  const int outm = 8 * khalf;
  const int outc = n0 + mrow;
  const float bb1 = b1[outc];
#pragma unroll
  for (int g = 0; g < 8; ++g) {
    float h = acc[g] + bb1;
    Ht[(outm + g) * HSTRIDE + outc] = h > 0.f ? h : 0.f;
  }
  __syncthreads();

  v8f acc2 = {};
  const float* hrow = Ht + mrow * HSTRIDE + khalf * 2;
  const float* w2row = W2 + (size_t)(n0 + mrow) * D + khalf * 2;
#pragma unroll 8
  for (int k0 = 0; k0 < D; k0 += 4) {
    v2f a = *(const v2f*)(hrow + k0);
    v2f b = *(const v2f*)(w2row + k0);
    acc2 = wmma4(a, b, acc2);
  }
  const float bb2 = b2[outc];
  float* srow = S + (size_t)(r0 + outm) * D + outc;
#pragma unroll
  for (int g = 0; g < 8; ++g) srow[(size_t)g * D] = acc2[g] + bb2;
}

__device__ __forceinline__ unsigned encOrd(float f) {
  unsigned u = __float_as_uint(f);
  return (u & 0x80000000u) ? ~u : (u | 0x80000000u);
}

// ---- edge kernels: one wave per edge; lane l owns columns [4l, 4l+4) ----
__global__ void edge_segmax_kernel(const int* __restrict__ head,
                                   const int* __restrict__ tail,
                                   const float* __restrict__ scores,
                                   unsigned* __restrict__ m, int tailVisit) {
  int e = blockIdx.x * 8 + (threadIdx.x >> 5);
  if (e >= NE) return;
  int h = head[e];
  if (h >= NV) return;
  int t = tail[e];
  if ((t < NV) != (tailVisit != 0)) return;
  int c = (threadIdx.x & 31) * 4;
  const float4 s = *(const float4*)(scores + (size_t)t * D + c);
  unsigned* mr = m + (size_t)h * D + c;
  atomicMax(mr + 0, encOrd(s.x));
  atomicMax(mr + 1, encOrd(s.y));
  atomicMax(mr + 2, encOrd(s.z));
  atomicMax(mr + 3, encOrd(s.w));
}

__global__ void decode_max_kernel(unsigned* __restrict__ m, int n) {
  int i = blockIdx.x * 256 + threadIdx.x;
  if (i >= n) return;
  unsigned e = m[i];
  float f;
  if (e == ENC_NEGINF) f = 0.f;  // empty segment -> 0 (matches isfinite fixup)
  else if (e & 0x80000000u) f = __uint_as_float(e & 0x7FFFFFFFu);
  else f = __uint_as_float(~e);
  ((float*)m)[i] = f;
}

// Fused: den[h] += e ; agg[h] += e * vals[t]   (att division deferred per-node)
__global__ void edge_accum_kernel(const int* __restrict__ head,
                                  const int* __restrict__ tail,
                                  const float* __restrict__ scores,
                                  const float* __restrict__ m,
                                  const float* __restrict__ vals,
                                  float* __restrict__ den,
                                  float* __restrict__ agg, int tailVisit) {
  int e = blockIdx.x * 8 + (threadIdx.x >> 5);
  if (e >= NE) return;
  int h = head[e];
  if (h >= NV) return;
  int t = tail[e];
  if ((t < NV) != (tailVisit != 0)) return;
  int c = (threadIdx.x & 31) * 4;
  const float4 s = *(const float4*)(scores + (size_t)t * D + c);
  const float4 mm = *(const float4*)(m + (size_t)h * D + c);
  const float4 v = *(const float4*)(vals + (size_t)t * D + c);
  float e0 = expf(s.x - mm.x);
  float e1 = expf(s.y - mm.y);
  float e2 = expf(s.z - mm.z);
  float e3 = expf(s.w - mm.w);
  float* dr = den + (size_t)h * D + c;
  float* ar = agg + (size_t)h * D + c;
  atomicAdd(dr + 0, e0);
  atomicAdd(dr + 1, e1);
  atomicAdd(dr + 2, e2);
  atomicAdd(dr + 3, e3);
  atomicAdd(ar + 0, e0 * v.x);
  atomicAdd(ar + 1, e1 * v.y);
  atomicAdd(ar + 2, e2 * v.z);
  atomicAdd(ar + 3, e3 * v.w);
}

// agg2 = (agg1 / den) * time_emb[row]
__global__ void divide_time_kernel(float* __restrict__ agg,
                                   const float* __restrict__ den,
                                   const float* __restrict__ te) {
  int i = blockIdx.x * 256 + threadIdx.x;
  if (i >= NV * D) return;
  float dd = den[i];
  dd = (dd == 0.f) ? 1.f : dd;
  agg[i] = agg[i] / dd * te[i >> 7];
}

__global__ void divide_kernel(float* __restrict__ out,
                              const float* __restrict__ den, int n) {
  int i = blockIdx.x * 256 + threadIdx.x;
  if (i >= n) return;
  float dd = den[i];
  dd = (dd == 0.f) ? 1.f : dd;
  out[i] /= dd;
}

// Offsets: union of iv/vi/vv masks == "head < NV" -> one max region for visits;
// ccs heads -> max; icd heads -> min (values are all >= 0 so raw bits order).
__global__ void edge_off_kernel(const int* __restrict__ head,
                                const int* __restrict__ tail,
                                const float* __restrict__ offs,
                                unsigned* __restrict__ ob) {
  int e = blockIdx.x * 8 + (threadIdx.x >> 5);
  if (e >= NE) return;
  int h = head[e];
  int t = tail[e];
  int c = (threadIdx.x & 31) * 4;
  const float4 o4 = *(const float4*)(offs + (size_t)t * D + c);
  unsigned* obr = ob + (size_t)h * D + c;
  if (h < NV + NC) {
    atomicMax(obr + 0, __float_as_uint(o4.x));
    atomicMax(obr + 1, __float_as_uint(o4.y));
    atomicMax(obr + 2, __float_as_uint(o4.z));
    atomicMax(obr + 3, __float_as_uint(o4.w));
  } else {
    atomicMin(obr + 0, __float_as_uint(o4.x));
    atomicMin(obr + 1, __float_as_uint(o4.y));
    atomicMin(obr + 2, __float_as_uint(o4.z));
    atomicMin(obr + 3, __float_as_uint(o4.w));
  }
}

__global__ void off_finalize_kernel(unsigned* __restrict__ ob) {
  int i = blockIdx.x * 256 + threadIdx.x;
  if (i >= NP * D) return;
  int row = i >> 7;
  unsigned u = ob[i];
  float f;
  if (row >= NN) f = 0.f;
  else if (row >= NV + NC) f = (u == 0x7F800000u) ? 0.f : __uint_as_float(u);
  else f = __uint_as_float(u);
  ((float*)ob)[i] = fmaxf(f, 0.f);
}

__global__ void init_kernel(const float* __restrict__ ve, const float* __restrict__ vo,
                            const float* __restrict__ ce, const float* __restrict__ co,
                            const float* __restrict__ ie, const float* __restrict__ io,
                            float* __restrict__ embs, float* __restrict__ offs) {
  int idx = blockIdx.x * 256 + threadIdx.x;
  if (idx >= NP * D) return;
  int row = idx >> 7, c = idx & 127;
  float e = 0.f, o = 0.f;
  if (row < NV) {
    e = ve[idx];
    o = vo[idx];
  } else if (row < NV + NC) {
    int r = row - NV;
    e = ce[r * D + c];
    o = co[r * D + c];
  } else if (row < NN) {
    int r = row - NV - NC;
    e = ie[r * D + c];
    o = io[r * D + c];
  }
  embs[idx] = e;
  offs[idx] = o > 0.f ? o : 0.f;
}

__global__ void fill_u32_kernel(unsigned* __restrict__ p, unsigned v, int n) {
  int i = blockIdx.x * 256 + threadIdx.x;
  if (i < n) p[i] = v;
}

__global__ void __launch_bounds__(1024)
time_softmax_kernel(const float* __restrict__ vt, const float* __restrict__ tw1,
                    const float* __restrict__ tb1, const float* __restrict__ tw2,
                    const float* __restrict__ tb2, float* __restrict__ te) {
  __shared__ float red[1024];
  int tid = threadIdx.x;
  float w1 = tw1[0], B1 = tb1[0], w2 = tw2[0], B2 = tb2[0];
  float lm = -3.4e38f;
  for (int i = tid; i < NV; i += 1024) {
    float x = (1.0f / vt[i]) * w1 + B1;
    x = x > 0.f ? x : 0.f;
    lm = fmaxf(lm, x * w2 + B2);
  }
  red[tid] = lm;
  __syncthreads();
  for (int o = 512; o > 0; o >>= 1) {
    if (tid < o) red[tid] = fmaxf(red[tid], red[tid + o]);
    __syncthreads();
  }
  float mx = red[0];
  __syncthreads();
  float ls = 0.f;
  for (int i = tid; i < NV; i += 1024) {
    float x = (1.0f / vt[i]) * w1 + B1;
    x = x > 0.f ? x : 0.f;
    ls += expf(x * w2 + B2 - mx);
  }
  red[tid] = ls;
  __syncthreads();
  for (int o = 512; o > 0; o >>= 1) {
    if (tid < o) red[tid] += red[tid + o];
    __syncthreads();
  }
  float inv = 1.f / red[0];
  for (int i = tid; i < NV; i += 1024) {
    float x = (1.0f / vt[i]) * w1 + B1;
    x = x > 0.f ? x : 0.f;
    te[i] = expf(x * w2 + B2 - mx) * inv;
  }
}

__global__ void normalize_kernel(float* __restrict__ v) {
  int row = blockIdx.x * 8 + (threadIdx.x >> 5);
  if (row >= NP) return;
  int lane = threadIdx.x & 31;
  float* r = v + (size_t)row * D;
  float s = 0.f;
#pragma unroll
  for (int c = lane; c < D; c += 32) {
    float x = r[c];
    s += x * x;
  }
#pragma unroll
  for (int o = 16; o > 0; o >>= 1) s += __shfl_down(s, o, 32);
  float nrm = sqrtf(__shfl(s, 0, 32));
  float inv = 1.f / fmaxf(nrm, 1e-12f);
#pragma unroll
  for (int c = lane; c < D; c += 32) r[c] *= inv;
}

extern "C" void kernel_launch(void* const* d_in, const int* in_sizes, int n_in,
                              void* d_out, int out_size, void* d_ws, size_t ws_size,
                              hipStream_t stream) {
  (void)in_sizes; (void)n_in; (void)out_size; (void)ws_size;
  const float* ve = (const float*)d_in[0];
  const float* vo = (const float*)d_in[1];
  const float* ce = (const float*)d_in[2];
  const float* co = (const float*)d_in[3];
  const float* ie = (const float*)d_in[4];
  const float* io = (const float*)d_in[5];
  const float* vt = (const float*)d_in[6];
  const float* W1 = (const float*)d_in[7];
  const float* b1 = (const float*)d_in[8];
  const float* W2 = (const float*)d_in[9];
  const float* b2 = (const float*)d_in[10];
  const float* tw1 = (const float*)d_in[11];
  const float* tb1 = (const float*)d_in[12];
  const float* tw2 = (const float*)d_in[13];
  const float* tb2 = (const float*)d_in[14];
  const int* ei = (const int*)d_in[15];
  const int* head = ei;
  const int* tail = ei + NE;

  float* ws = (float*)d_ws;
  size_t o = 0;
  float* embsA = ws + o; o += (size_t)NP * D;
  float* embsB = ws + o; o += (size_t)NP * D;
  float* offA  = ws + o; o += (size_t)NP * D;
  float* offB  = ws + o; o += (size_t)NP * D;
  float* scores = ws + o; o += (size_t)NP * D;
  float* mbuf  = ws + o; o += (size_t)NV * D;
  float* den   = ws + o; o += (size_t)NV * D;
  float* agg1  = ws + o; o += (size_t)NV * D;
  float* te    = ws + o; o += NV;

  dim3 b256(256);
  int gNPD = (NP * D + 255) / 256;
  int gNVD = (NV * D + 255) / 256;
  int gE = (NE + 7) / 8;

  init_kernel<<<gNPD, b256, 0, stream>>>(ve, vo, ce, co, ie, io, embsA, offA);
  time_softmax_kernel<<<1, 1024, 0, stream>>>(vt, tw1, tb1, tw2, tb2, te);

  float* cur = embsA; float* nxt = embsB;
  float* curO = offA; float* nxtO = offB;
  for (int hop = 0; hop < 2; ++hop) {
    // --- center net A (edges visit<-nonvisit), per-node score precompute ---
    mlp2_kernel<<<NP / 16, b256, 0, stream>>>(cur, W1, b1, W2, b2, scores);
    fill_u32_kernel<<<gNVD, b256, 0, stream>>>((unsigned*)mbuf, ENC_NEGINF, NV * D);
    fill_u32_kernel<<<gNVD, b256, 0, stream>>>((unsigned*)den, 0u, NV * D);
    fill_u32_kernel<<<gNVD, b256, 0, stream>>>((unsigned*)agg1, 0u, NV * D);
    edge_segmax_kernel<<<gE, b256, 0, stream>>>(head, tail, scores, (unsigned*)mbuf, 0);
    decode_max_kernel<<<gNVD, b256, 0, stream>>>((unsigned*)mbuf, NV * D);
    edge_accum_kernel<<<gE, b256, 0, stream>>>(head, tail, scores, mbuf, cur, den, agg1, 0);
    divide_time_kernel<<<gNVD, b256, 0, stream>>>(agg1, den, te);  // agg2 in place

    // --- center net B (edges visit<-visit) on agg2 ---
    mlp2_kernel<<<NV / 16, b256, 0, stream>>>(agg1, W1, b1, W2, b2, scores);
    fill_u32_kernel<<<gNVD, b256, 0, stream>>>((unsigned*)mbuf, ENC_NEGINF, NV * D);
    fill_u32_kernel<<<gNVD, b256, 0, stream>>>((unsigned*)den, 0u, NV * D);
    fill_u32_kernel<<<gNPD, b256, 0, stream>>>((unsigned*)nxt, 0u, NP * D);
    edge_segmax_kernel<<<gE, b256, 0, stream>>>(head, tail, scores, (unsigned*)mbuf, 1);
    decode_max_kernel<<<gNVD, b256, 0, stream>>>((unsigned*)mbuf, NV * D);
    edge_accum_kernel<<<gE, b256, 0, stream>>>(head, tail, scores, mbuf, agg1, den, nxt, 1);
    divide_kernel<<<gNVD, b256, 0, stream>>>(nxt, den, NV * D);
    normalize_kernel<<<(NP + 7) / 8, b256, 0, stream>>>(nxt);

    // --- offsets ---
    fill_u32_kernel<<<gNPD, b256, 0, stream>>>((unsigned*)nxtO, 0u, NP * D);
    fill_u32_kernel<<<(NI * D + 255) / 256, b256, 0, stream>>>(
        (unsigned*)nxtO + (size_t)(NV + NC) * D, 0x7F800000u, NI * D);
    edge_off_kernel<<<gE, b256, 0, stream>>>(head, tail, curO, (unsigned*)nxtO);
    off_finalize_kernel<<<gNPD, b256, 0, stream>>>((unsigned*)nxtO);

    float* t1 = cur; cur = nxt; nxt = t1;
    float* t2 = curO; curO = nxtO; nxtO = t2;
  }

  hipMemcpyAsync(d_out, cur, (size_t)NV * D * sizeof(float),
                 hipMemcpyDeviceToDevice, stream);
  hipMemcpyAsync((float*)d_out + (size_t)NV * D, curO, (size_t)NV * D * sizeof(float),
                 hipMemcpyDeviceToDevice, stream);
}